// EM_online_83528523973074
// MI455X (gfx1250) — compile-verified
//
#include <hip/hip_runtime.h>
#include <cstdint>

#define NCOLS 500000
#define RDIM 32
#define BSZ 1024
#define ETAF 0.01f

// ---- gfx1250 async global<->LDS copy availability (device pass only) ----
#if defined(__HIP_DEVICE_COMPILE__) && defined(__has_builtin)
#if __has_builtin(__builtin_amdgcn_global_load_async_to_lds_b128) && \
    __has_builtin(__builtin_amdgcn_global_store_async_from_lds_b128) && \
    __has_builtin(__builtin_amdgcn_s_wait_asynccnt)
#define USE_ASYNC_COPY 1
#endif
#endif
#ifndef USE_ASYNC_COPY
#define USE_ASYNC_COPY 0
#endif

#if USE_ASYNC_COPY
// The b128 async builtins take pointers to a 16-byte int vector in AS(1)/AS(3).
typedef int v4i __attribute__((vector_size(16)));
// Generic pointer to a __shared__ object carries the LDS byte offset in its
// low 32 bits (SHARED_BASE aperture occupies the high bits), so truncation
// yields a valid AS(3) pointer. Global generic pointers are numerically AS(1).
#define AS1V4(p) ((__attribute__((address_space(1))) v4i*)(unsigned long long)(p))
#define AS3V4(p) ((__attribute__((address_space(3))) v4i*)(unsigned)(unsigned long long)(p))
#endif

// -------- Kernel 1: bulk copy v[d] -> out[d] (bandwidth-dominant) --------
__global__ void k_copy(const float4* __restrict__ src, float4* __restrict__ dst, int n4) {
#if USE_ASYNC_COPY
  __shared__ float4 buf[8][256];
  const int tid = threadIdx.x;
  const long long stride = (long long)gridDim.x * 2048;
  for (long long base = (long long)blockIdx.x * 2048; base < n4; base += stride) {
#pragma unroll
    for (int s = 0; s < 8; ++s) {
      long long i = base + s * 256 + tid;
      if (i < n4) {
        __builtin_amdgcn_global_load_async_to_lds_b128(
            AS1V4(src + i), AS3V4(&buf[s][tid]), 0, 0);
      }
    }
    __builtin_amdgcn_s_wait_asynccnt(0);
#pragma unroll
    for (int s = 0; s < 8; ++s) {
      long long i = base + s * 256 + tid;
      if (i < n4) {
        __builtin_amdgcn_global_store_async_from_lds_b128(
            AS1V4(dst + i), AS3V4(&buf[s][tid]), 0, 0);
      }
    }
    __builtin_amdgcn_s_wait_asynccnt(0);  // buffers reusable next iteration
  }
#else
  long long i = (long long)blockIdx.x * blockDim.x + threadIdx.x;
  const long long stride = (long long)gridDim.x * blockDim.x;
  for (; i < n4; i += stride) dst[i] = src[i];
#endif
}

// -------- Kernel 2: per-b stats. One wave32 per b, lane = r. --------
__global__ void k_stats(const int* __restrict__ entries, const int* __restrict__ vals,
                        const float* __restrict__ v0, const float* __restrict__ v1,
                        const float* __restrict__ v2, const float* __restrict__ L,
                        float* __restrict__ Q, float* __restrict__ WI,
                        float* __restrict__ KI, float* __restrict__ PP) {
  const int r = threadIdx.x & 31;
  const int b = blockIdx.x * (blockDim.x >> 5) + (threadIdx.x >> 5);
  if (b >= BSZ) return;
  const int c0 = entries[b * 3 + 0];
  const int c1 = entries[b * 3 + 1];
  const int c2 = entries[b * 3 + 2];
  float g0 = v0[(size_t)c0 * RDIM + r];
  float g1 = v1[(size_t)c1 * RDIM + r];
  float g2 = v2[(size_t)c2 * RDIM + r];
  float lr = L[r];
  float a0 = g0, a1 = g1, a2 = g2, ph = lr * g0 * g1 * g2;
#pragma unroll
  for (int off = 16; off >= 1; off >>= 1) {
    a0 *= __shfl_xor(a0, off, 32);
    a1 *= __shfl_xor(a1, off, 32);
    a2 *= __shfl_xor(a2, off, 32);
    ph += __shfl_xor(ph, off, 32);
  }
  if (r == 0) {
    const float P = a0 * a1 * a2;
    Q[0 * BSZ + b] = P / a0;   // Q[b,d] = P / A_d
    Q[1 * BSZ + b] = P / a1;
    Q[2 * BSZ + b] = P / a2;
    WI[b] = 0.5f / ph * tanhf(0.5f * ph);
    KI[b] = (float)vals[b] - 0.5f;
    PP[b] = P;
  }
}

// -------- Kernel 3: duplicate links: prev occurrence + is-last flag --------
__global__ void k_links(const int* __restrict__ entries,
                        int* __restrict__ prev, int* __restrict__ last) {
  __shared__ int ent[BSZ * 3];
  for (int i = threadIdx.x; i < BSZ * 3; i += blockDim.x) ent[i] = entries[i];
  __syncthreads();
  const int t = blockIdx.x * blockDim.x + threadIdx.x;
  if (t >= 3 * BSZ) return;
  const int d = t / BSZ;
  const int b = t - d * BSZ;
  const int col = ent[b * 3 + d];
  int p = -1;
  for (int b2 = b - 1; b2 >= 0; --b2)
    if (ent[b2 * 3 + d] == col) { p = b2; break; }
  int isl = 1;
  for (int b2 = b + 1; b2 < BSZ; ++b2)
    if (ent[b2 * 3 + d] == col) { isl = 0; break; }
  prev[t] = p;
  last[t] = isl;
}

// -------- Kernel 4: chain-walk EMA closed form, scatter final rows --------
__global__ void k_scatter(const int* __restrict__ entries, const float* __restrict__ L,
                          const float* __restrict__ S0, const float* __restrict__ S1,
                          const float* __restrict__ S2, const float* __restrict__ T0,
                          const float* __restrict__ T1, const float* __restrict__ T2,
                          const float* __restrict__ Q, const float* __restrict__ WI,
                          const float* __restrict__ KI, const float* __restrict__ PP,
                          const int* __restrict__ prev, const int* __restrict__ last,
                          float* __restrict__ out) {
  const int w = blockIdx.x * (blockDim.x >> 5) + (threadIdx.x >> 5);
  const int r = threadIdx.x & 31;
  if (w >= 3 * BSZ) return;
  const int d = w / BSZ;
  const int b = w - d * BSZ;
  if (!last[w]) return;  // only the final occurrence of (d,col) writes
  const int col = entries[b * 3 + d];
  const float lr = L[r];
  float sl = lr;  // sum(L) via wave32 butterfly
#pragma unroll
  for (int off = 16; off >= 1; off >>= 1) sl += __shfl_xor(sl, off, 32);

  // S' = (1-eta)S + eta*upd, occurrences u1..um (um most recent):
  // final = (1-eta)^m * S0 + eta * sum_j (1-eta)^(m-j) * u_j.
  // Walk backwards from the most recent occurrence.
  float accS = 0.f, accT = 0.f, decay = 1.f;
  int bb = b;
  while (true) {
    const float q  = Q[d * BSZ + bb];
    const float wi = WI[bb];
    const float ki = KI[bb];
    const float P  = PP[bb];
    const float c    = q * lr;             // c[b,d,r] = (P/A_d) * L[r]
    const float updS = c * c * wi;
    const float dv   = P * (sl - lr);      // dvec[b,r] = P * (sumL - L[r])
    const float updT = (ki - dv * wi) * c;
    accS += decay * ETAF * updS;
    accT += decay * ETAF * updT;
    decay *= (1.0f - ETAF);
    const int pb = prev[d * BSZ + bb];
    if (pb < 0) break;
    bb = pb;
  }
  const float* Sb = (d == 0) ? S0 : (d == 1 ? S1 : S2);
  const float* Tb = (d == 0) ? T0 : (d == 1 ? T1 : T2);
  const size_t idx = (size_t)col * RDIM + r;
  const float Sfin = accS + decay * Sb[idx];
  const float Tfin = accT + decay * Tb[idx];
  out[(size_t)d * NCOLS * RDIM + idx] = Sfin / Tfin;
}

extern "C" void kernel_launch(void* const* d_in, const int* in_sizes, int n_in,
                              void* d_out, int out_size, void* d_ws, size_t ws_size,
                              hipStream_t stream) {
  (void)in_sizes; (void)n_in; (void)out_size; (void)ws_size;
  const int*   entries = (const int*)d_in[0];
  const int*   vals    = (const int*)d_in[1];
  const float* v0 = (const float*)d_in[2];
  const float* v1 = (const float*)d_in[3];
  const float* v2 = (const float*)d_in[4];
  const float* L  = (const float*)d_in[5];
  const float* S0 = (const float*)d_in[6];
  const float* S1 = (const float*)d_in[7];
  const float* S2 = (const float*)d_in[8];
  const float* T0 = (const float*)d_in[9];
  const float* T1 = (const float*)d_in[10];
  const float* T2 = (const float*)d_in[11];
  float* out = (float*)d_out;

  // workspace carve (48 KB)
  float* wsf = (float*)d_ws;
  float* Q    = wsf;             // 3*B
  float* WI   = wsf + 3 * BSZ;   // B
  float* KI   = wsf + 4 * BSZ;   // B
  float* PP   = wsf + 5 * BSZ;   // B
  int*   prev = (int*)(wsf + 6 * BSZ);  // 3*B
  int*   last = prev + 3 * BSZ;         // 3*B

  // 1) bulk copies (dominant: 384 MB total traffic)
  const int n4 = NCOLS * RDIM / 4;              // 4,000,000 float4 per array
  const int copyBlocks = (n4 + 2047) / 2048;    // one 8-slot batch per block
  const float* vsrc[3] = {v0, v1, v2};
  for (int d = 0; d < 3; ++d) {
    k_copy<<<copyBlocks, 256, 0, stream>>>(
        (const float4*)vsrc[d], (float4*)(out + (size_t)d * NCOLS * RDIM), n4);
  }

  // 2) per-b stats: 1024 waves
  k_stats<<<BSZ / 8, 256, 0, stream>>>(entries, vals, v0, v1, v2, L, Q, WI, KI, PP);

  // 3) duplicate-occurrence links
  k_links<<<(3 * BSZ + 255) / 256, 256, 0, stream>>>(entries, prev, last);

  // 4) scatter final rows (runs after copy in stream order)
  k_scatter<<<(3 * BSZ * 32 + 255) / 256, 256, 0, stream>>>(
      entries, L, S0, S1, S2, T0, T1, T2, Q, WI, KI, PP, prev, last, out);
}